// GraphAttention_63376537420062
// MI455X (gfx1250) — compile-verified
//
#include <hip/hip_runtime.h>

typedef __attribute__((ext_vector_type(2))) float v2f;
typedef __attribute__((ext_vector_type(8))) float v8f;

#define NHEADS 8

// Order-preserving float <-> uint encoding so atomicMax(uint) implements
// a float segment-max. All finite floats encode to > 0, so init value 0
// behaves as -inf identity.
__device__ __forceinline__ unsigned enc_f32(float f) {
  unsigned u = __float_as_uint(f);
  return (u & 0x80000000u) ? ~u : (u | 0x80000000u);
}
__device__ __forceinline__ float dec_f32(unsigned e) {
  return (e & 0x80000000u) ? __uint_as_float(e & 0x7FFFFFFFu)
                           : __uint_as_float(~e);
}

__global__ void ga_init(float* __restrict__ out, size_t out_n,
                        unsigned* __restrict__ menc, float* __restrict__ denom,
                        size_t nh) {
  size_t i = (size_t)blockIdx.x * blockDim.x + threadIdx.x;
  size_t stride = (size_t)gridDim.x * blockDim.x;
  for (size_t t = i; t < out_n; t += stride) out[t] = 0.0f;
  for (size_t t = i; t < nh; t += stride) {
    menc[t] = 0u;
    denom[t] = 0.0f;
  }
}

// Stage 1: e[E,H] = <khat, qhat[dst]>/sqrt(128); segment-max via atomicMax.
// One wave handles 2 edges x 8 heads = 16 (edge,head) pairs.
// Pair m is both A-row m and B-col m; 4 chained V_WMMA_F32_16X16X4_F32 cover
// the D=16 reduction; the wanted dots sit on the accumulator diagonal.
// Diagonal residency (C/D layout, ISA 7.12.2): lanes 0-7 hold C[m,m] for
// m=lane at c[lane]; lanes 24-31 hold C[m,m] for m=lane-16 at c[lane-24].
// Those lanes' (edge, head, node) from the load phase are already correct,
// so extraction is a lane-local 8-way select — no cross-lane ops needed.
__global__ void ga_scores(const float* __restrict__ q0, const float* __restrict__ q1,
                          const float* __restrict__ k0, const float* __restrict__ k1,
                          const long long* __restrict__ dst,
                          float* __restrict__ escore, unsigned* __restrict__ menc,
                          long long E) {
  const float scale = 0.088388347648318447f;  // 1/sqrt(128)
  const unsigned lane = threadIdx.x & 31u;
  const long long wid =
      (long long)blockIdx.x * (blockDim.x >> 5) + (threadIdx.x >> 5);
  const long long e0 = wid * 2;
  if (e0 >= E) return;

  if (e0 + 1 < E) {  // wave-uniform: EXEC stays all-1s for WMMA
    const unsigned p = lane & 15u;   // pair id = A row = B col
    const unsigned hi = lane >> 4;   // 0: K={4c,4c+1}, 1: K={4c+2,4c+3}
    const long long edge = e0 + (p >> 3);
    const unsigned head = p & 7u;
    const long long node = dst[edge];

    // khat[e,h,d]: d<4 -> k0[e*32+4h+d]; d>=4 -> k1[e*96+12h+(d-4)].
    const float* kr0 = k0 + edge * 32 + head * 4 + 2 * hi;
    const float* kr1 = k1 + edge * 96 + head * 12 + 2 * hi;
    const float* qr0 = q0 + node * 32 + head * 4 + 2 * hi;
    const float* qr1 = q1 + node * 96 + head * 12 + 2 * hi;

    v2f a0 = *(const v2f*)kr0;        // K chunk 0 (d 0..3)
    v2f a1 = *(const v2f*)kr1;        // K chunk 1 (d 4..7)
    v2f a2 = *(const v2f*)(kr1 + 4);  // K chunk 2 (d 8..11)
    v2f a3 = *(const v2f*)(kr1 + 8);  // K chunk 3 (d 12..15)
    v2f b0 = *(const v2f*)qr0;
    v2f b1 = *(const v2f*)qr1;
    v2f b2 = *(const v2f*)(qr1 + 4);
    v2f b3 = *(const v2f*)(qr1 + 8);

    v8f c = {0.f, 0.f, 0.f, 0.f, 0.f, 0.f, 0.f, 0.f};
    c = __builtin_amdgcn_wmma_f32_16x16x4_f32(false, a0, false, b0, (short)0, c, false, false);
    c = __builtin_amdgcn_wmma_f32_16x16x4_f32(false, a1, false, b1, (short)0, c, false, false);
    c = __builtin_amdgcn_wmma_f32_16x16x4_f32(false, a2, false, b2, (short)0, c, false, false);
    c = __builtin_amdgcn_wmma_f32_16x16x4_f32(false, a3, false, b3, (short)0, c, false, false);

    // Lane-local diagonal pick: val = c[lane & 7].
    const unsigned sel = lane & 7u;
    float val = c[0];
    if (sel == 1) val = c[1];
    if (sel == 2) val = c[2];
    if (sel == 3) val = c[3];
    if (sel == 4) val = c[4];
    if (sel == 5) val = c[5];
    if (sel == 6) val = c[6];
    if (sel == 7) val = c[7];

    // Lanes 0-7 own pairs 0-7 (edge e0); lanes 24-31 own pairs 8-15 (e0+1).
    if (lane < 8u || lane >= 24u) {
      float e_val = val * scale;
      escore[edge * NHEADS + head] = e_val;
      atomicMax(&menc[node * NHEADS + head], enc_f32(e_val));
    }
  } else {
    // Scalar tail: single remaining edge, lanes 0..7 = heads.
    if (lane < NHEADS) {
      long long edge = e0;
      long long node = dst[edge];
      const float* kr0 = k0 + edge * 32 + lane * 4;
      const float* kr1 = k1 + edge * 96 + lane * 12;
      const float* qr0 = q0 + node * 32 + lane * 4;
      const float* qr1 = q1 + node * 96 + lane * 12;
      float s = 0.0f;
#pragma unroll
      for (int d = 0; d < 4; ++d) s += kr0[d] * qr0[d];
#pragma unroll
      for (int d = 0; d < 12; ++d) s += kr1[d] * qr1[d];
      s *= scale;
      escore[edge * NHEADS + lane] = s;
      atomicMax(&menc[node * NHEADS + lane], enc_f32(s));
    }
  }
}

// Stage 2: ex = exp(e - m[dst]) in place; denom += ex per (node, head).
__global__ void ga_softmax(float* __restrict__ escore,
                           const long long* __restrict__ dst,
                           const unsigned* __restrict__ menc,
                           float* __restrict__ denom, long long total) {
  long long i = (long long)blockIdx.x * blockDim.x + threadIdx.x;
  if (i >= total) return;
  long long edge = i >> 3;
  unsigned h = (unsigned)(i & 7);
  long long node = dst[edge];
  float m = dec_f32(menc[node * NHEADS + h]);
  float ex = __expf(escore[i] - m);
  escore[i] = ex;
  atomicAdd(&denom[node * NHEADS + h], ex);
}

// Stage 3: out += (ex/denom[dst]) * v, 128 value elements per edge.
// d_out layout: [0, N*32) = out0 flat, [N*32, N*128) = out1 flat.
__global__ void ga_aggregate(const float* __restrict__ v0, const float* __restrict__ v1,
                             const long long* __restrict__ dst,
                             const float* __restrict__ ex, const float* __restrict__ denom,
                             float* __restrict__ out, long long E, long long N) {
  long long i = (long long)blockIdx.x * blockDim.x + threadIdx.x;
  long long total = E * 128;
  if (i >= total) return;
  long long edge = i >> 7;
  unsigned t = (unsigned)(i & 127);
  long long node = dst[edge];
  unsigned h;
  float val;
  long long slot;
  if (t < 32) {
    h = t >> 2;  // channel/4
    val = v0[edge * 32 + t];
    slot = node * 32 + t;
  } else {
    unsigned u = t - 32;          // index into [32 ch][3 comps] flat
    h = (u / 3) >> 2;             // channel = u/3; head = channel/4
    val = v1[edge * 96 + u];
    slot = N * 32 + node * 96 + u;
  }
  float a = ex[edge * NHEADS + h] / denom[node * NHEADS + h];
  atomicAdd(&out[slot], a * val);
}

extern "C" void kernel_launch(void* const* d_in, const int* in_sizes, int n_in,
                              void* d_out, int out_size, void* d_ws, size_t ws_size,
                              hipStream_t stream) {
  const float* q0 = (const float*)d_in[0];
  const float* q1 = (const float*)d_in[1];
  const float* k0 = (const float*)d_in[2];
  const float* k1 = (const float*)d_in[3];
  const float* v0 = (const float*)d_in[4];
  const float* v1 = (const float*)d_in[5];
  const long long* dst = (const long long*)d_in[6];

  const long long E = in_sizes[6];        // edge count
  const long long N = in_sizes[0] / 32;   // node count
  float* out = (float*)d_out;

  // Workspace: [E*H f32 scores->ex | N*H u32 encoded max | N*H f32 denom]
  char* ws = (char*)d_ws;
  float* escore = (float*)ws;
  unsigned* menc = (unsigned*)(ws + (size_t)E * NHEADS * sizeof(float));
  float* denom = (float*)(ws + (size_t)E * NHEADS * sizeof(float) +
                          (size_t)N * NHEADS * sizeof(unsigned));

  const size_t nh = (size_t)N * NHEADS;
  ga_init<<<dim3(4096), dim3(256), 0, stream>>>(out, (size_t)out_size, menc,
                                                denom, nh);

  {
    long long waves = (E + 1) / 2;          // 2 edges per wave
    long long blocks = (waves + 7) / 8;     // 8 waves per 256-thread block
    ga_scores<<<dim3((unsigned)blocks), dim3(256), 0, stream>>>(
        q0, q1, k0, k1, dst, escore, menc, E);
  }
  {
    long long total = E * NHEADS;
    long long blocks = (total + 255) / 256;
    ga_softmax<<<dim3((unsigned)blocks), dim3(256), 0, stream>>>(
        escore, dst, menc, denom, total);
  }
  {
    long long total = E * 128;
    long long blocks = (total + 255) / 256;
    ga_aggregate<<<dim3((unsigned)blocks), dim3(256), 0, stream>>>(
        v0, v1, dst, escore, denom, out, E, N);
  }
}